// LinearTimeMMDLoss_73821897883949
// MI455X (gfx1250) — compile-verified
//
#include <hip/hip_runtime.h>

// LinearTimeMMDLoss on MI455X (gfx1250, wave32).
// Memory-bound (256 MB @ 23.3 TB/s ~= 11 us). Distances via diag(D.D^T)
// accumulated with V_WMMA_F32_16X16X4_F32 (fp32, matches reference).
// Raw 16x32 tiles of xo/xe/yo/ye are streamed into LDS with
// GLOBAL_LOAD_ASYNC_TO_LDS_B128 (double-buffered, ASYNCcnt-pipelined);
// diffs are formed on the 2-wide WMMA operands in registers.

typedef float v2f __attribute__((ext_vector_type(2)));
typedef float v8f __attribute__((ext_vector_type(8)));
typedef int v4i __attribute__((vector_size(4 * sizeof(int))));  // builtin's pointee type

#define WAVES_PER_BLOCK 2
#define DIMS 512
#define KC 32
#define NCHUNK (DIMS / KC)
#define LDS_STRIDE 36            // 144B rows: 16B aligned, 36*r mod 64 distinct
#define TILE_F (16 * LDS_STRIDE) // floats per 16xKC tile
#define BUF_F (4 * TILE_F)       // xo, xe, yo, ye

#if __has_builtin(__builtin_amdgcn_global_load_async_to_lds_b128)
#define HAVE_ASYNC_LDS 1
#else
#define HAVE_ASYNC_LDS 0
#endif

__device__ __forceinline__ float4 ld4(const float* __restrict__ p) {
    return *(const float4*)p;
}
__device__ __forceinline__ void st4(float* p, float4 v) { *(float4*)p = v; }

__device__ __forceinline__ v8f gram_acc(v2f d, v8f c) {
    // A layout (16x4 f32): lane L<16 -> row L,(K0,K1); lane L+16 -> row L,(K2,K3).
    // Same register pattern serves B = A^T, so C += D.D^T; diag = ||row||^2.
    return __builtin_amdgcn_wmma_f32_16x16x4_f32(
        false, d, false, d, (short)0, c, false, false);
}

__device__ __forceinline__ void wait_async_le16() {
#if HAVE_ASYNC_LDS
#if __has_builtin(__builtin_amdgcn_s_wait_asynccnt)
    __builtin_amdgcn_s_wait_asynccnt(16);
#else
    asm volatile("s_wait_asynccnt 16" ::: "memory");
#endif
    asm volatile("" ::: "memory");
#else
    __syncthreads();
#endif
}
__device__ __forceinline__ void wait_async_le0() {
#if HAVE_ASYNC_LDS
#if __has_builtin(__builtin_amdgcn_s_wait_asynccnt)
    __builtin_amdgcn_s_wait_asynccnt(0);
#else
    asm volatile("s_wait_asynccnt 0" ::: "memory");
#endif
    asm volatile("" ::: "memory");
#else
    __syncthreads();
#endif
}

__device__ __forceinline__ void async_cp_b128(const float* g, float* l) {
#if HAVE_ASYNC_LDS
    __builtin_amdgcn_global_load_async_to_lds_b128((v4i*)g, (v4i*)l, 0, 0);
#else
    st4(l, ld4(g));
#endif
}

// Stage raw 16xKC tiles xo/xe/yo/ye for chunk k0 into buf (per-wave region).
// 16 x b128 per wave: lane -> (sub-row lane>>3, 4-col group (lane&7)*4).
__device__ __forceinline__ void issue_copies(const float* __restrict__ src,
                                             const float* __restrict__ tgt,
                                             long base_row, int k0,
                                             float* buf, int srow, int scol) {
#pragma unroll
    for (int rg = 0; rg < 4; ++rg) {
        const int row = rg * 4 + srow;
        const long go = (base_row + 2 * (long)row) * DIMS + k0 + scol;
        const int lo = row * LDS_STRIDE + scol;
        async_cp_b128(src + go,        buf + 0 * TILE_F + lo);
        async_cp_b128(src + go + DIMS, buf + 1 * TILE_F + lo);
        async_cp_b128(tgt + go,        buf + 2 * TILE_F + lo);
        async_cp_b128(tgt + go + DIMS, buf + 3 * TILE_F + lo);
    }
}

__device__ __forceinline__ void compute_chunk(const float* buf, int mrow, int koff,
                                              v8f acc[4]) {
#pragma unroll
    for (int k4 = 0; k4 < KC; k4 += 4) {
        const int o = mrow * LDS_STRIDE + k4 + koff;
        const v2f xo = *(const v2f*)(buf + 0 * TILE_F + o);  // ds_load_b64
        const v2f xe = *(const v2f*)(buf + 1 * TILE_F + o);
        const v2f yo = *(const v2f*)(buf + 2 * TILE_F + o);
        const v2f ye = *(const v2f*)(buf + 3 * TILE_F + o);
        acc[0] = gram_acc(xo - xe, acc[0]);
        acc[1] = gram_acc(yo - ye, acc[1]);
        acc[2] = gram_acc(xo - ye, acc[2]);
        acc[3] = gram_acc(xe - yo, acc[3]);
    }
}

__global__ __launch_bounds__(WAVES_PER_BLOCK * 32)
void mmd_tile_kernel(const float* __restrict__ src,
                     const float* __restrict__ tgt,
                     float* __restrict__ partial) {
    __shared__ float lds[WAVES_PER_BLOCK][2][BUF_F];  // double-buffered raw tiles

    const int lane = threadIdx.x & 31;
    const int wave = threadIdx.x >> 5;
    const int tile = blockIdx.x * WAVES_PER_BLOCK + wave;
    const long base_row = (long)tile * 32;

    const int srow = lane >> 3;
    const int scol = (lane & 7) * 4;
    const int mrow = lane & 15;
    const int koff = (lane >> 4) * 2;

    float* buf[2] = { &lds[wave][0][0], &lds[wave][1][0] };

    v8f acc[4] = {};  // Gram accumulators: xx, yy, xy, yx

    issue_copies(src, tgt, base_row, 0, buf[0], srow, scol);
#pragma unroll 2
    for (int c = 0; c < NCHUNK; ++c) {
        const int cur = c & 1;
        if (c + 1 < NCHUNK) {
            issue_copies(src, tgt, base_row, (c + 1) * KC, buf[cur ^ 1], srow, scol);
            wait_async_le16();  // chunk c's 16 copies done (in-order), c+1 in flight
        } else {
            wait_async_le0();
        }
        compute_chunk(buf[cur], mrow, koff, acc);
    }

    // ---- diagonal extraction via LDS ----
    // C/D f32 16x16: VGPR v, lanes 0-15 -> (M=v, N=lane); lanes 16-31 -> (M=v+8).
    float* wlds = buf[0];
    __syncthreads();
    {
        const int n = lane & 15;
        const int madd = (lane >> 4) * 8;
#pragma unroll
        for (int t = 0; t < 4; ++t) {
#pragma unroll
            for (int v = 0; v < 8; ++v) {
                wlds[t * 256 + (v + madd) * 16 + n] = acc[t][v];
            }
        }
    }
    __syncthreads();

    if (lane < 16) {
        const float dxx = wlds[0 * 256 + lane * 17];
        const float dyy = wlds[1 * 256 + lane * 17];
        const float dxy = wlds[2 * 256 + lane * 17];
        const float dyx = wlds[3 * 256 + lane * 17];
        // bw = (sum/4) / 2^(5//2) = sum/16; bws_k = bw*2^k + eps
        const float bw = (dxx + dyy + dxy + dyx) * 0.0625f;
        float hp = 0.0f;
        float mult = 1.0f;
#pragma unroll
        for (int kk = 0; kk < 5; ++kk) {
            const float b = bw * mult + 1e-6f;
            const float rb = 1.0f / b;
            hp += __expf(-dxx * rb) + __expf(-dyy * rb)
                - __expf(-dxy * rb) - __expf(-dyx * rb);
            mult *= 2.0f;
        }
        wlds[1024 + lane] = hp;
    }
    __syncthreads();

    if (lane == 0) {
        float s = 0.0f;
#pragma unroll
        for (int i = 0; i < 16; ++i) s += wlds[1024 + i];
        partial[tile] = s;
    }
}

// Deterministic fixed-order reduction of per-tile partials.
__global__ __launch_bounds__(256)
void mmd_reduce_kernel(const float* __restrict__ partial, float* __restrict__ out,
                       int n, float inv_m2) {
    __shared__ float sm[256];
    float s = 0.0f;
    for (int i = threadIdx.x; i < n; i += 256) s += partial[i];
    sm[threadIdx.x] = s;
    __syncthreads();
    for (int off = 128; off > 0; off >>= 1) {
        if ((int)threadIdx.x < off) sm[threadIdx.x] += sm[threadIdx.x + off];
        __syncthreads();
    }
    if (threadIdx.x == 0) out[0] = sm[0] * inv_m2;
}

extern "C" void kernel_launch(void* const* d_in, const int* in_sizes, int n_in,
                              void* d_out, int out_size, void* d_ws, size_t ws_size,
                              hipStream_t stream) {
    const float* src = (const float*)d_in[0];
    const float* tgt = (const float*)d_in[1];
    float* out = (float*)d_out;
    float* ws = (float*)d_ws;

    const int m = in_sizes[0] / DIMS;            // 65536
    const int m2 = m / 2;                        // 32768 pairs
    const int tiles = m2 / 16;                   // 2048 (16 pairs per wave)
    const int blocks = tiles / WAVES_PER_BLOCK;  // 1024 blocks x 2 waves

    mmd_tile_kernel<<<blocks, WAVES_PER_BLOCK * 32, 0, stream>>>(src, tgt, ws);
    mmd_reduce_kernel<<<1, 256, 0, stream>>>(ws, out, tiles, 1.0f / (float)m2);
}